// word_choser_52561809768553
// MI455X (gfx1250) — compile-verified
//
#include <hip/hip_runtime.h>
#include <hip/hip_bf16.h>
#include <math.h>

// word_choser on MI455X (gfx1250, wave32).
//
// Algebraic collapse of the reference:
//   row-mean commutes with right-matmul  -> [4096,2048]x[2048,2048] GEMM
//   becomes colsum(hiddens) @ dim_up;  all GCN node rows identical ->
//   softmax uniform -> attention pool returns that row (att_w unused).
// Remaining work: 4 vector@matrix GEMVs (V_WMMA_F32_16X16X4_F32, B-stream
// is the only VMEM traffic, A staged in LDS), 8192 row-dots for the LSTM
// gates (float4 global loads + LDS-resident small vectors + shuffle
// reduce), pointwise LSTM.  ~547 MB total traffic -> bandwidth bound,
// ~23 us at 23.3 TB/s.

#define S_LEN   4096
#define D_DIM   2048
#define NTOK    32000
#define INPDIM  (2 * D_DIM + 1)   // 4097
#define NNODES  (S_LEN + 1)       // 4097

typedef float v2f __attribute__((ext_vector_type(2)));
typedef float v8f __attribute__((ext_vector_type(8)));

__global__ void wc_zero_kernel(float* __restrict__ p, int n) {
    int i = blockIdx.x * blockDim.x + threadIdx.x;
    if (i < n) p[i] = 0.0f;
}

// csum[d] = sum_s hiddens[s][d]; each thread owns 4 contiguous columns
// (float4 loads), 256-row slab partial sums + atomicAdd.
__global__ void wc_colsum_kernel(const float* __restrict__ hiddens,
                                 float* __restrict__ csum) {
    int d4 = blockIdx.x * blockDim.x + threadIdx.x;   // 0..511
    int s0 = blockIdx.y * 256;
    const float4* p = (const float4*)hiddens + (size_t)s0 * (D_DIM / 4) + d4;
    float4 acc = make_float4(0.f, 0.f, 0.f, 0.f);
#pragma unroll 8
    for (int s = 0; s < 256; ++s) {
        float4 v = *p;
        acc.x += v.x; acc.y += v.y; acc.z += v.z; acc.w += v.w;
        p += D_DIM / 4;
    }
    atomicAdd(&csum[4 * d4 + 0], acc.x);
    atomicAdd(&csum[4 * d4 + 1], acc.y);
    atomicAdd(&csum[4 * d4 + 2], acc.z);
    atomicAdd(&csum[4 * d4 + 3], acc.w);
}

// out[n] = epilogue( (sum_k vec[k]*M[k*N+n] + add[n]) * scale )
//
// One wave per 16-wide n-chunk.  vec staged in LDS (A-operand reads use the
// DS pipe; VMEM carries only the B matrix).  K-step 16: 4 independent
// accumulators -> 4 v_wmma_f32_16x16x4_f32 per iteration, no RAW chain.
// A = vec chunk broadcast into all 16 rows (only D row 0 consumed).
//   A layout (16x4 f32): lane L, half h=L>>4: VGPR0=K(2h), VGPR1=K(2h+1)
//   B layout (4x16 f32): lane L, half h:     VGPR0=row 2h, VGPR1=row 2h+1
//   D layout: row 0 in c[0] of lanes 0..15 (N = n0 + lane).
// Grids are sized exactly (waves == N/16): no wave exits before the
// barrier, EXEC is all-ones at every WMMA.
__global__ void wc_gemv_wmma_kernel(const float* __restrict__ vec,
                                    const float* __restrict__ M,
                                    const float* __restrict__ addvec,
                                    float* __restrict__ out,
                                    int K, int N, float scale, int do_relu) {
    __shared__ __align__(16) float s_vec[D_DIM];    // K <= 2048 always here

    for (int i = threadIdx.x; i < K; i += blockDim.x)
        s_vec[i] = vec[i];
    __syncthreads();

    int wave = (int)((blockIdx.x * blockDim.x + threadIdx.x) >> 5);
    int lane = (int)(threadIdx.x & 31);
    int n0 = wave * 16;
    if (n0 >= N) return;                            // wave-uniform (never hit)

    const int half = lane >> 4;
    const int nl   = lane & 15;
    const int k0   = half * 2;                      // VGPR0 K index in 4-tile
    const int k1   = half * 2 + 1;                  // VGPR1 K index in 4-tile

    const float* col = M + n0 + nl;

    v8f c0 = {}, c1 = {}, c2 = {}, c3 = {};

    for (int k = 0; k < K; k += 16) {
        if (k + 16 < K)
            __builtin_prefetch(&col[(size_t)(k + 16 + k0) * N], 0, 0);

        v2f a0, b0;
        a0.x = s_vec[k + k0];            a0.y = s_vec[k + k1];
        b0.x = col[(size_t)(k + k0) * N];
        b0.y = col[(size_t)(k + k1) * N];
        c0 = __builtin_amdgcn_wmma_f32_16x16x4_f32(
                 false, a0, false, b0, (short)0, c0, false, false);

        v2f a1, b1;
        a1.x = s_vec[k + 4 + k0];        a1.y = s_vec[k + 4 + k1];
        b1.x = col[(size_t)(k + 4 + k0) * N];
        b1.y = col[(size_t)(k + 4 + k1) * N];
        c1 = __builtin_amdgcn_wmma_f32_16x16x4_f32(
                 false, a1, false, b1, (short)0, c1, false, false);

        v2f a2, b2;
        a2.x = s_vec[k + 8 + k0];        a2.y = s_vec[k + 8 + k1];
        b2.x = col[(size_t)(k + 8 + k0) * N];
        b2.y = col[(size_t)(k + 8 + k1) * N];
        c2 = __builtin_amdgcn_wmma_f32_16x16x4_f32(
                 false, a2, false, b2, (short)0, c2, false, false);

        v2f a3, b3;
        a3.x = s_vec[k + 12 + k0];       a3.y = s_vec[k + 12 + k1];
        b3.x = col[(size_t)(k + 12 + k0) * N];
        b3.y = col[(size_t)(k + 12 + k1) * N];
        c3 = __builtin_amdgcn_wmma_f32_16x16x4_f32(
                 false, a3, false, b3, (short)0, c3, false, false);
    }

    if (half == 0) {                                // lanes 0..15: D row 0
        float r = (c0[0] + c1[0]) + (c2[0] + c3[0]);
        if (addvec) r += addvec[n0 + nl];
        r *= scale;
        if (do_relu) r = fmaxf(r, 0.0f);
        out[n0 + nl] = r;
    }
}

__global__ void wc_build_inp_kernel(const float* __restrict__ sen_emb,
                                    const int* __restrict__ pos_index,
                                    float* __restrict__ inp) {
    int i = blockIdx.x * blockDim.x + threadIdx.x;
    if (i < D_DIM) inp[i] = sen_emb[i];
    if (i == 0)    inp[2 * D_DIM] = (float)pos_index[0];
}

// gsum[g*D+h] = Wi[g][h]·inp + Wh[g][h]·cur_h + bi + bh
// One wave per row.  inp/cur_h staged in LDS (DS pipe); Wi body read as
// float4 after alignment peel (rows are 4097 floats -> base alignment
// varies by wave%4); Wh rows are 8KB-aligned -> straight float4.
__global__ void wc_lstm_gates_kernel(const float* __restrict__ Wi,
                                     const float* __restrict__ Wh,
                                     const float* __restrict__ bi,
                                     const float* __restrict__ bh,
                                     const float* __restrict__ inp,
                                     const float* __restrict__ cur_h,
                                     float* __restrict__ gsum) {
    __shared__ __align__(16) float s_inp[INPDIM];   // 4097
    __shared__ __align__(16) float s_h[D_DIM];      // 2048

    for (int i = threadIdx.x; i < INPDIM; i += blockDim.x) s_inp[i] = inp[i];
    for (int i = threadIdx.x; i < D_DIM;  i += blockDim.x) s_h[i]   = cur_h[i];
    __syncthreads();

    int wave = (int)((blockIdx.x * blockDim.x + threadIdx.x) >> 5);
    int lane = (int)(threadIdx.x & 31);
    if (wave >= 4 * D_DIM) return;                  // never hit (exact grid)

    float acc = 0.0f;

    // ---- Wi[g][h] . inp  (row length 4097, element misalignment wave&3) ----
    {
        const float* wi = Wi + (size_t)wave * INPDIM;
        const int pe   = (4 - (wave & 3)) & 3;      // peel to 16B alignment
        const int L    = INPDIM - pe;
        const int n4   = L >> 2;
        const int tail = L & 3;

        if (lane < pe)
            acc += wi[lane] * s_inp[lane];

        const float4* wf4 = (const float4*)(wi + pe);
        for (int j = lane; j < n4; j += 32) {
            float4 w = wf4[j];
            int base = pe + 4 * j;
            acc += w.x * s_inp[base + 0];
            acc += w.y * s_inp[base + 1];
            acc += w.z * s_inp[base + 2];
            acc += w.w * s_inp[base + 3];
        }

        if (lane < tail) {
            int base = pe + 4 * n4;
            acc += wi[base + lane] * s_inp[base + lane];
        }
    }

    // ---- Wh[g][h] . cur_h  (row length 2048, 8KB-aligned) ----
    {
        const float4* wh4 = (const float4*)(Wh + (size_t)wave * D_DIM);
        const float4* sh4 = (const float4*)s_h;
        for (int j = lane; j < D_DIM / 4; j += 32) {
            float4 w = wh4[j];
            float4 h = sh4[j];
            acc += w.x * h.x + w.y * h.y + w.z * h.z + w.w * h.w;
        }
    }

#pragma unroll
    for (int off = 16; off > 0; off >>= 1)
        acc += __shfl_down(acc, off, 32);

    if (lane == 0) gsum[wave] = acc + bi[wave] + bh[wave];
}

__device__ __forceinline__ float wc_sigmoid(float x) {
    return 1.0f / (1.0f + expf(-x));
}

__global__ void wc_lstm_pointwise_kernel(const float* __restrict__ gsum,
                                         const float* __restrict__ cur_cell,
                                         float* __restrict__ new_h) {
    int h = blockIdx.x * blockDim.x + threadIdx.x;
    if (h >= D_DIM) return;
    float i_g = wc_sigmoid(gsum[0 * D_DIM + h]);
    float f_g = wc_sigmoid(gsum[1 * D_DIM + h]);
    float g_g = tanhf     (gsum[2 * D_DIM + h]);
    float o_g = wc_sigmoid(gsum[3 * D_DIM + h]);
    float nc  = f_g * cur_cell[h] + i_g * g_g;
    new_h[h]  = o_g * tanhf(nc);
}

extern "C" void kernel_launch(void* const* d_in, const int* in_sizes, int n_in,
                              void* d_out, int out_size, void* d_ws, size_t ws_size,
                              hipStream_t stream) {
    (void)in_sizes; (void)n_in; (void)out_size; (void)ws_size;

    const float* sen_emb   = (const float*)d_in[0];
    const float* hiddens   = (const float*)d_in[1];
    const int*   pos_index = (const int*)  d_in[2];
    const float* dim_up    = (const float*)d_in[3];
    const float* dim_down  = (const float*)d_in[4];
    const float* dim_out   = (const float*)d_in[5];
    const float* W_gcn     = (const float*)d_in[6];
    /* d_in[7] = att_w: unused (softmax of identical scores is uniform). */
    const float* Wi        = (const float*)d_in[8];
    const float* Wh        = (const float*)d_in[9];
    const float* bi        = (const float*)d_in[10];
    const float* bh        = (const float*)d_in[11];
    const float* cur_h     = (const float*)d_in[12];
    const float* cur_cell  = (const float*)d_in[13];

    float* out = (float*)d_out;

    float* csum  = (float*)d_ws;          // 2048
    float* mmsg  = csum + D_DIM;          // 2048
    float* vgcn  = mmsg + D_DIM;          // 2048
    float* inp   = vgcn + D_DIM;          // 4097 (padded 4104)
    float* gsum  = inp + 4104;            // 8192
    float* new_h = gsum + 4 * D_DIM;      // 2048

    const int gemvD_blocks = (D_DIM / 16) * 32 / 256;   // 16
    const int gemvO_blocks = (NTOK  / 16) * 32 / 256;   // 250

    // 1) csum = colsum(hiddens)
    wc_zero_kernel<<<(D_DIM + 255) / 256, 256, 0, stream>>>(csum, D_DIM);
    wc_colsum_kernel<<<dim3((D_DIM / 4) / 256, S_LEN / 256), 256, 0, stream>>>(
        hiddens, csum);

    // 2) mean_msg = (csum @ dim_up + cur_h) / (S+1)
    wc_gemv_wmma_kernel<<<gemvD_blocks, 256, 0, stream>>>(
        csum, dim_up, cur_h, mmsg, D_DIM, D_DIM, 1.0f / (float)NNODES, 0);

    // 3) v = relu(mean_msg @ W_gcn)   (== attention-pooled node embedding)
    wc_gemv_wmma_kernel<<<gemvD_blocks, 256, 0, stream>>>(
        mmsg, W_gcn, nullptr, vgcn, D_DIM, D_DIM, 1.0f, 1);

    // 4) the_inp = cat(sen_emb, v @ dim_down, pos)
    wc_build_inp_kernel<<<(D_DIM + 255) / 256, 256, 0, stream>>>(
        sen_emb, pos_index, inp);
    wc_gemv_wmma_kernel<<<gemvD_blocks, 256, 0, stream>>>(
        vgcn, dim_down, nullptr, inp + D_DIM, D_DIM, D_DIM, 1.0f, 0);

    // 5) gate pre-activations: 8192 rows, one wave each
    wc_lstm_gates_kernel<<<(4 * D_DIM * 32) / 256, 256, 0, stream>>>(
        Wi, Wh, bi, bh, inp, cur_h, gsum);

    // 6) pointwise LSTM -> new_h
    wc_lstm_pointwise_kernel<<<(D_DIM + 255) / 256, 256, 0, stream>>>(
        gsum, cur_cell, new_h);

    // 7) out = new_h @ dim_out   [1, 32000]
    wc_gemv_wmma_kernel<<<gemvO_blocks, 256, 0, stream>>>(
        new_h, dim_out, nullptr, out, D_DIM, NTOK, 1.0f, 0);
}